// VoxelToPixel_90417651515813
// MI455X (gfx1250) — compile-verified
//
#include <hip/hip_runtime.h>
#include <hip/hip_bf16.h>

typedef __attribute__((ext_vector_type(2))) float v2f;
typedef __attribute__((ext_vector_type(8))) float v8f;

#define IMG_W 512
#define IMG_H 512
#define HW (IMG_W * IMG_H)
#define EPS 1e-6f

// ---------------------------------------------------------------------------
// Kernel 0: init packed z-buffer to all-ones (== +inf depth, no winner)
// ---------------------------------------------------------------------------
__global__ void vtp_init_zbuf(unsigned long long* __restrict__ zb, int n) {
    int i = blockIdx.x * blockDim.x + threadIdx.x;
    if (i < n) zb[i] = ~0ULL;
}

// ---------------------------------------------------------------------------
// Kernel 1: project 16 points/wave with V_WMMA_F32_16X16X4_F32 and scatter
// packed (depth_bits<<32)|(0x7FFFFFFF - idx) via one 64-bit atomicMin.
//  A (16x4, MxK): A[m][k] = P[m][k]  (m<3, else 0)
//  B (4x16, KxN): B[k][n] = coords_h[point n][k]
//  D (16x16)    : lane n (0..15), VGPR m -> p_m of point n  (no shuffles!)
// Striping per ISA 7.12.2: lanes 0-15 hold K=0,1 (vgpr0,1); lanes 16-31 K=2,3.
// Loads are branch-free: clamped addresses + v_cndmask value selects, so the
// two global_load_b32 per lane issue unconditionally (no saveexec regions).
// ---------------------------------------------------------------------------
__global__ void vtp_project_scatter(const int* __restrict__ coords,
                                    const float* __restrict__ P,
                                    unsigned long long* __restrict__ zb,
                                    int N) {
    const int lane = threadIdx.x & 31;
    const int q    = lane & 15;          // A: row m ; B: point-in-chunk n
    const int kb   = (lane >> 4) << 1;   // K base: 0 (lanes 0-15) or 2 (16-31)
    const bool loK = (kb == 0);          // loop-invariant lane mask

    // A matrix = projection matrix (wave-invariant, loaded once)
    v2f a;
    a[0] = (q < 3) ? P[q * 4 + kb]     : 0.0f;
    a[1] = (q < 3) ? P[q * 4 + kb + 1] : 0.0f;

    const int wavesPerBlock = blockDim.x >> 5;
    const long long wave   = (long long)blockIdx.x * wavesPerBlock + (threadIdx.x >> 5);
    const long long nWaves = (long long)gridDim.x * wavesPerBlock;
    const long long nChunks = (N + 15) / 16;

    for (long long c = wave; c < nChunks; c += nWaves) {
        const int base = (int)(c << 4);
        const int idx  = base + q;
        const bool inN = (idx < N);

        // Branch-free loads: clamp address, select value.
        const int cidx = idx < N ? idx : (N - 1);
        const int off  = cidx * 3;
        const float l0 = (float)coords[off + kb];               // x (lo) / z (hi)
        const float l1 = (float)coords[off + 1 + (kb >> 1)];    // y (lo) / z dummy (hi)

        v2f b;
        b[0] = inN ? l0 : 0.0f;
        b[1] = inN ? (loK ? l1 : 1.0f) : 0.0f;   // K=1 -> y ; K=3 -> homogeneous 1

        v8f acc = {};
        // D = A x B + 0   (EXEC all ones: loop bounds are wave-uniform)
        v8f d = __builtin_amdgcn_wmma_f32_16x16x4_f32(
            false, a, false, b, (short)0, acc, false, false);

        if (lane < 16 && inN) {
            const float px    = d[0];
            const float py    = d[1];
            const float depth = d[2];
            const float safe  = (depth > EPS) ? depth : 1.0f;
            const float fu = floorf(px / safe);
            const float fv = floorf(py / safe);
            if (depth > EPS && fu >= 0.0f && fu < (float)IMG_W &&
                fv >= 0.0f && fv < (float)IMG_H) {
                const int u = (int)fu, v = (int)fv;
                const int flat = v * IMG_W + u;
                // depth > 0 => float bits order-preserving as unsigned.
                // low 32 bits: 0x7FFFFFFF - idx => min(packed) picks min depth,
                // then max index (matches reference tie-break). Order-free.
                const unsigned long long packed =
                    ((unsigned long long)__float_as_uint(depth) << 32) |
                    (unsigned long long)(0x7FFFFFFFu - (unsigned)idx);
                atomicMin(&zb[flat], packed);
            }
        }
    }
}

// ---------------------------------------------------------------------------
// Kernel 2: resolve + gather. 16 threads per pixel, one float4 each
// (64 floats = 256 B contiguous per feature row -> global_load/store_b128).
// Writes every element of d_out (features region then inv-depth region).
// ---------------------------------------------------------------------------
__global__ void vtp_resolve_gather(const float* __restrict__ feats,
                                   const unsigned long long* __restrict__ zb,
                                   float* __restrict__ out) {
    const int tid = blockIdx.x * blockDim.x + threadIdx.x;
    const int pix = tid >> 4;
    const int t   = tid & 15;
    if (pix >= HW) return;

    const unsigned long long packed = zb[pix];  // 2 MB buffer: L2 resident
    const bool occ = (packed != ~0ULL);

    float4 val = make_float4(0.0f, 0.0f, 0.0f, 0.0f);
    if (occ) {
        const unsigned idx = 0x7FFFFFFFu - (unsigned)(packed & 0xFFFFFFFFu);
        const float4* f4 = (const float4*)feats;
        val = f4[(size_t)idx * 16 + t];
    }
    float4* out4 = (float4*)out;
    out4[(size_t)pix * 16 + t] = val;

    if (t == 0) {
        float invd = 0.0f;
        if (occ) {
            const float depth = __uint_as_float((unsigned)(packed >> 32));
            invd = 1.0f / depth;
        }
        out[(size_t)HW * 64 + pix] = invd;
    }
}

// ---------------------------------------------------------------------------
extern "C" void kernel_launch(void* const* d_in, const int* in_sizes, int n_in,
                              void* d_out, int out_size, void* d_ws, size_t ws_size,
                              hipStream_t stream) {
    const float* features = (const float*)d_in[0];   // (N, 64) f32
    const int*   coords   = (const int*)d_in[1];     // (N, 3)  i32
    const float* projmat  = (const float*)d_in[2];   // (3, 4)  f32
    float* out = (float*)d_out;

    const int N = in_sizes[1] / 3;
    unsigned long long* zb = (unsigned long long*)d_ws;  // HW * 8 bytes = 2 MB

    // Pass 0: clear packed z-buffer
    vtp_init_zbuf<<<(HW + 255) / 256, 256, 0, stream>>>(zb, HW);

    // Pass 1: WMMA projection + packed 64-bit atomicMin scatter
    vtp_project_scatter<<<2048, 256, 0, stream>>>(coords, projmat, zb, N);

    // Pass 2: resolve winners, gather feature rows, write inv-depth
    vtp_resolve_gather<<<(HW * 16 + 255) / 256, 256, 0, stream>>>(features, zb, out);
}